// GCNLayer_14886356648858
// MI455X (gfx1250) — compile-verified
//
#include <hip/hip_runtime.h>

typedef __attribute__((ext_vector_type(2))) float v2f;
typedef __attribute__((ext_vector_type(4))) float v4f;
typedef __attribute__((ext_vector_type(8))) float v8f;

#define N_NODES 100000
#define D_IN    256
#define D_OUT   256

// -------------------------------------------------------------------------
// Phase 0 — GEMM: h = x @ W^T via V_WMMA_F32_16X16X4_F32 (exact fp32).
// One wave per 16x64 strip; A fragment reused across 4 B tiles.
// -------------------------------------------------------------------------
__global__ __launch_bounds__(256)
void gcn_gemm_wmma(const float* __restrict__ x,
                   const float* __restrict__ W,
                   float* __restrict__ h) {
    const int lane = threadIdx.x & 31;
    const int wave = threadIdx.x >> 5;
    const int tile = blockIdx.x * (blockDim.x >> 5) + wave;
    const int tm    = tile >> 2;
    const int strip = tile & 3;
    if (tm * 16 >= N_NODES) return;

    const int m    = tm * 16 + (lane & 15);
    const int kb   = (lane >> 4) << 1;
    const int ncol = strip * 64 + (lane & 15);

    const float* __restrict__ xrow = x + (size_t)m * D_IN;

    v8f c0 = {}, c1 = {}, c2 = {}, c3 = {};

    #pragma unroll 4
    for (int k = 0; k < D_IN; k += 4) {
        v2f a  = *(const v2f*)(xrow + k + kb);
        v2f b0 = *(const v2f*)(W + (size_t)(ncol     ) * D_IN + k + kb);
        v2f b1 = *(const v2f*)(W + (size_t)(ncol + 16) * D_IN + k + kb);
        v2f b2 = *(const v2f*)(W + (size_t)(ncol + 32) * D_IN + k + kb);
        v2f b3 = *(const v2f*)(W + (size_t)(ncol + 48) * D_IN + k + kb);

        c0 = __builtin_amdgcn_wmma_f32_16x16x4_f32(false, a, false, b0,
                                                   (short)0, c0, false, false);
        c1 = __builtin_amdgcn_wmma_f32_16x16x4_f32(false, a, false, b1,
                                                   (short)0, c1, false, false);
        c2 = __builtin_amdgcn_wmma_f32_16x16x4_f32(false, a, false, b2,
                                                   (short)0, c2, false, false);
        c3 = __builtin_amdgcn_wmma_f32_16x16x4_f32(false, a, false, b3,
                                                   (short)0, c3, false, false);
    }

    const int rbase = tm * 16 + ((lane >> 4) << 3);
    #pragma unroll
    for (int v = 0; v < 8; ++v) {
        float* hr = h + (size_t)(rbase + v) * D_OUT + ncol;
        hr[ 0] = c0[v];
        hr[16] = c1[v];
        hr[32] = c2[v];
        hr[48] = c3[v];
    }
}

// -------------------------------------------------------------------------
// Phase 1 — histogram: cnt[r] = #edges with row==r (u32 atomics, cheap).
// -------------------------------------------------------------------------
__global__ __launch_bounds__(256)
void gcn_hist(const int* __restrict__ edge_row,
              unsigned* __restrict__ cnt, int n_edges) {
    const int e = blockIdx.x * blockDim.x + threadIdx.x;
    if (e < n_edges) atomicAdd(&cnt[edge_row[e]], 1u);
}

// -------------------------------------------------------------------------
// Phase 2 — exclusive scan of cnt -> rowStart[N+1], cursor[N].
// Single 1024-thread block; each thread owns ~98 rows; LDS Hillis-Steele
// across the 1024 per-thread partial sums.
// -------------------------------------------------------------------------
__global__ __launch_bounds__(1024)
void gcn_scan(const unsigned* __restrict__ cnt,
              unsigned* __restrict__ rowStart,
              unsigned* __restrict__ cursor) {
    __shared__ unsigned part[1024];
    const int t   = threadIdx.x;
    const int per = (N_NODES + 1023) / 1024;
    const int lo  = t * per;
    const int hi  = (lo + per < N_NODES) ? lo + per : N_NODES;

    unsigned s = 0;
    for (int i = lo; i < hi; ++i) s += cnt[i];
    part[t] = s;
    __syncthreads();

    for (int off = 1; off < 1024; off <<= 1) {
        unsigned v = (t >= off) ? part[t - off] : 0u;
        __syncthreads();
        part[t] += v;
        __syncthreads();
    }

    unsigned run = (t == 0) ? 0u : part[t - 1];   // exclusive prefix
    for (int i = lo; i < hi; ++i) {
        rowStart[i] = run;
        cursor[i]   = run;
        run += cnt[i];
    }
    if (t == 1023) rowStart[N_NODES] = part[1023];
}

// -------------------------------------------------------------------------
// Phase 3 — counting-sort scatter: permute (val, col) into row-sorted order.
// -------------------------------------------------------------------------
__global__ __launch_bounds__(256)
void gcn_scatter(const int*   __restrict__ edge_row,
                 const int*   __restrict__ edge_col,
                 const float* __restrict__ edge_val,
                 unsigned*    __restrict__ cursor,
                 float* __restrict__ sVal,
                 int*   __restrict__ sCol, int n_edges) {
    const int e = blockIdx.x * blockDim.x + threadIdx.x;
    if (e >= n_edges) return;
    const unsigned p = atomicAdd(&cursor[edge_row[e]], 1u);
    sVal[p] = edge_val[e];
    sCol[p] = edge_col[e];
}

// -------------------------------------------------------------------------
// Phase 4 — wave-per-row segment reduction: out[r] = sum val*h[col].
// 8 accumulator VGPRs/lane (256 floats/wave); b128 gathers from L2-resident
// h; no atomics; every output row written (no pre-zero needed).
// -------------------------------------------------------------------------
__global__ __launch_bounds__(256)
void gcn_row_accum(const float* __restrict__ h,
                   const float* __restrict__ sVal,
                   const int*   __restrict__ sCol,
                   const unsigned* __restrict__ rowStart,
                   float* __restrict__ out) {
    const int lane = threadIdx.x & 31;
    int r = blockIdx.x * (blockDim.x >> 5) + (threadIdx.x >> 5);
    if (r >= N_NODES) return;                  // wave-uniform guard
    r = __builtin_amdgcn_readfirstlane(r);     // scalar index loads

    const unsigned e0 = rowStart[r];
    const unsigned e1 = rowStart[r + 1];

    const int d0 = lane * 4;
    const int d1 = 128 + lane * 4;
    v4f acc0 = {}, acc1 = {};

    for (unsigned e = e0; e < e1; ++e) {
        const float v = sVal[e];
        const int   c = sCol[e];
        const float* __restrict__ hs = h + (size_t)c * D_OUT;
        v4f h0 = *(const v4f*)(hs + d0);
        v4f h1 = *(const v4f*)(hs + d1);
        v4f vv = {v, v, v, v};
        acc0 += vv * h0;
        acc1 += vv * h1;
    }

    float* __restrict__ od = out + (size_t)r * D_OUT;
    *(v4f*)(od + d0) = acc0;
    *(v4f*)(od + d1) = acc1;
}

// -------------------------------------------------------------------------
// Fallback (workspace too small): wave-per-edge fp32 atomic scatter.
// -------------------------------------------------------------------------
__global__ __launch_bounds__(256)
void gcn_spmm_atomic(const float* __restrict__ h,
                     const float* __restrict__ edge_val,
                     const int*   __restrict__ edge_row,
                     const int*   __restrict__ edge_col,
                     float* __restrict__ out, int n_edges) {
    const int lane = threadIdx.x & 31;
    int e = blockIdx.x * (blockDim.x >> 5) + (threadIdx.x >> 5);
    if (e >= n_edges) return;
    e = __builtin_amdgcn_readfirstlane(e);

    const int   r = edge_row[e];
    const int   c = edge_col[e];
    const float v = edge_val[e];
    const float* __restrict__ hs = h   + (size_t)c * D_OUT;
    float*       __restrict__ od = out + (size_t)r * D_OUT;

    #pragma unroll
    for (int i = 0; i < 8; ++i) {            // lane-contiguous atomics
        const int d = i * 32 + lane;
        unsafeAtomicAdd(od + d, v * hs[d]);
    }
}

extern "C" void kernel_launch(void* const* d_in, const int* in_sizes, int n_in,
                              void* d_out, int out_size, void* d_ws, size_t ws_size,
                              hipStream_t stream) {
    const float* x        = (const float*)d_in[0];
    const float* W        = (const float*)d_in[1];
    const float* edge_val = (const float*)d_in[2];
    const int*   edge_row = (const int*)d_in[3];
    const int*   edge_col = (const int*)d_in[4];
    float*       out      = (float*)d_out;
    const int n_edges = in_sizes[2];

    // ---- workspace layout (256B-aligned slices) ----
    size_t off = 0;
    auto carve = [&](size_t bytes) {
        char* p = (char*)d_ws + off;
        off += (bytes + 255) & ~(size_t)255;
        return (void*)p;
    };
    float*    h        = (float*)   carve((size_t)N_NODES * D_OUT * 4);
    unsigned* cnt      = (unsigned*)carve((size_t)N_NODES * 4);
    unsigned* rowStart = (unsigned*)carve(((size_t)N_NODES + 1) * 4);
    unsigned* cursor   = (unsigned*)carve((size_t)N_NODES * 4);
    float*    sVal     = (float*)   carve((size_t)n_edges * 4);
    int*      sCol     = (int*)     carve((size_t)n_edges * 4);
    const bool csr_ok = (off <= ws_size);

    // ---- GEMM into workspace ----
    const int n_tiles = (N_NODES / 16) * (D_OUT / 64);
    gcn_gemm_wmma<<<dim3((n_tiles + 7) / 8), dim3(256), 0, stream>>>(x, W, h);

    if (csr_ok) {
        hipMemsetAsync(cnt, 0, (size_t)N_NODES * 4, stream);
        gcn_hist<<<dim3((n_edges + 255) / 256), dim3(256), 0, stream>>>(
            edge_row, cnt, n_edges);
        gcn_scan<<<dim3(1), dim3(1024), 0, stream>>>(cnt, rowStart, cursor);
        gcn_scatter<<<dim3((n_edges + 255) / 256), dim3(256), 0, stream>>>(
            edge_row, edge_col, edge_val, cursor, sVal, sCol, n_edges);
        gcn_row_accum<<<dim3((N_NODES + 7) / 8), dim3(256), 0, stream>>>(
            h, sVal, sCol, rowStart, out);
    } else {
        hipMemsetAsync(d_out, 0, (size_t)out_size * sizeof(float), stream);
        gcn_spmm_atomic<<<dim3((n_edges + 7) / 8), dim3(256), 0, stream>>>(
            h, edge_val, edge_row, edge_col, out, n_edges);
    }
}